// GCN5_MN_COORD_67980742361105
// MI455X (gfx1250) — compile-verified
//
#include <hip/hip_runtime.h>
#include <hip/hip_bf16.h>
#include <math.h>

typedef float v2f __attribute__((ext_vector_type(2)));
typedef float v8f __attribute__((ext_vector_type(8)));

// ---------------------------------------------------------------- utilities
// vectorized zero: each thread clears 4 floats (b128 store), scalar tail
__global__ void gcn_zero(float* __restrict__ p, int n) {
    int i4 = (blockIdx.x * blockDim.x + threadIdx.x) * 4;
    if (i4 + 3 < n) {
        *reinterpret_cast<float4*>(p + i4) = make_float4(0.f, 0.f, 0.f, 0.f);
    } else if (i4 < n) {
        for (int k = i4; k < n; ++k) p[k] = 0.0f;
    }
}

// degrees as f32 (matches segment_sum of ones)
__global__ void gcn_degree(const int* __restrict__ src, const int* __restrict__ dst,
                           float* __restrict__ in_deg, float* __restrict__ out_deg, int nE) {
    int e = blockIdx.x * blockDim.x + threadIdx.x;
    if (e >= nE) return;
    atomicAdd(out_deg + src[e], 1.0f);
    atomicAdd(in_deg  + dst[e], 1.0f);
}

// norms + scaled layer-0 features hs0 = [in_deg, x, y, 0] * norm_s
__global__ void gcn_feat0(const float* __restrict__ cx, const float* __restrict__ cy,
                          const float* __restrict__ in_deg, const float* __restrict__ out_deg,
                          float* __restrict__ norm_s, float* __restrict__ norm_d,
                          float* __restrict__ hs0, int nN) {
    int n = blockIdx.x * blockDim.x + threadIdx.x;
    if (n >= nN) return;
    float id = in_deg[n], od = out_deg[n];
    float ns = 1.0f / sqrtf(fmaxf(od, 1.0f));
    float nd = 1.0f / sqrtf(fmaxf(id, 1.0f));
    norm_s[n] = ns;
    norm_d[n] = nd;
    float4 v = make_float4(id * ns, cx[n] * ns, cy[n] * ns, 0.0f);
    *reinterpret_cast<float4*>(hs0 + 4 * (size_t)n) = v;
}

// pad W1 (3x128) -> Wpad (4x128) with zero last row
__global__ void gcn_pad_w1(const float* __restrict__ W1, float* __restrict__ Wpad) {
    int i = blockIdx.x * blockDim.x + threadIdx.x;   // 0..511
    if (i >= 512) return;
    Wpad[i] = ((i >> 7) < 3) ? W1[i] : 0.0f;
}

// ---------------------------------------------------------------- scatters
// layer-1: 3-wide aggregation, one thread per edge
__global__ void gcn_scatter3(const float* __restrict__ hs0, const int* __restrict__ src,
                             const int* __restrict__ dst, float* __restrict__ agg4, int nE) {
    int e = blockIdx.x * blockDim.x + threadIdx.x;
    if (e >= nE) return;
    int s = src[e], d = dst[e];
    float4 v = *reinterpret_cast<const float4*>(hs0 + 4 * (size_t)s);
    float* ap = agg4 + 4 * (size_t)d;
    atomicAdd(ap + 0, v.x);
    atomicAdd(ap + 1, v.y);
    atomicAdd(ap + 2, v.z);
}

// 128-wide aggregation: 32 lanes per edge; edge index is wave-uniform, so pin it
// to an SGPR with readfirstlane (scalar src/dst address math, broadcast loads).
__global__ void gcn_scatter128(const float* __restrict__ hs, const int* __restrict__ src,
                               const int* __restrict__ dst, float* __restrict__ agg, int nE) {
    int g = blockIdx.x * blockDim.x + threadIdx.x;
    int e = __builtin_amdgcn_readfirstlane(g >> 5);   // uniform across the wave
    if (e >= nE) return;
    int c = (g & 31) * 4;
    int s = src[e], d = dst[e];
    float4 v = *reinterpret_cast<const float4*>(hs + (size_t)s * 128 + c);
    float* ap = agg + (size_t)d * 128 + c;
    atomicAdd(ap + 0, v.x);
    atomicAdd(ap + 1, v.y);
    atomicAdd(ap + 2, v.z);
    atomicAdd(ap + 3, v.w);
}

// ---------------------------------------------------------------- WMMA GEMM
// h' = relu((Ain * norm_d) @ W + b), fused epilogue selected at compile time:
//   MODE 0: store hs' = h' * norm_s          (intermediate layers)
//   MODE 1: store h' to output + column sums (final layer)
// Block: 256 threads = 8 waves; wave w owns column tile [16w,16w+16); block does
// TPB row tiles of 16. B fragments live in VGPRs for the whole block.
template<int K, int TPB, int MODE>
__global__ __launch_bounds__(256)
void gcn_gemm_wmma(const float* __restrict__ Ain,     // [N,K]
                   const float* __restrict__ W,       // [K,128] row-major
                   const float* __restrict__ bias,    // [128]
                   const float* __restrict__ norm_d,  // [N]
                   const float* __restrict__ norm_s,  // [N]
                   float* __restrict__ hs_out,        // [N,128]   (MODE 0)
                   float* __restrict__ h_out,         // [N,128]   (MODE 1)
                   float* __restrict__ colsum,        // [128]     (MODE 1)
                   int nN) {
    const int lane = threadIdx.x & 31;
    const int wid  = threadIdx.x >> 5;
    const int half = lane >> 4;
    const int l15  = lane & 15;
    const int col  = wid * 16 + l15;

    // Preload B fragments: b = { W[4kk+2h][col], W[4kk+2h+1][col] }
    v2f bfrag[K / 4];
#pragma unroll
    for (int kk = 0; kk < K / 4; ++kk) {
        int kb = 4 * kk + 2 * half;
        bfrag[kk].x = W[kb * 128 + col];
        bfrag[kk].y = W[(kb + 1) * 128 + col];
    }
    const float bcol = bias[col];
    float colacc = 0.0f;

    const int tile0 = blockIdx.x * TPB;
    for (int t = 0; t < TPB; ++t) {
        const int rowBase = (tile0 + t) * 16;        // uniform -> EXEC stays full
        if (rowBase >= nN) break;
        int mRow = rowBase + l15;
        int mCl  = mRow < nN ? mRow : (nN - 1);      // clamp loads on tail tile
        const float ndm = norm_d[mCl];
        const float* arow = Ain + (size_t)mCl * K;

        v8f acc = {};
#pragma unroll
        for (int kk = 0; kk < K / 4; ++kk) {
            float2 av = *reinterpret_cast<const float2*>(arow + 4 * kk + 2 * half);
            v2f a;
            a.x = av.x * ndm;
            a.y = av.y * ndm;
            acc = __builtin_amdgcn_wmma_f32_16x16x4_f32(
                false, a, false, bfrag[kk], (short)0, acc, false, false);
        }

        // D layout: VGPR r -> row = rowBase + 8*half + r, column = col
        const int row0 = rowBase + 8 * half;
        if (rowBase + 16 <= nN) {                    // uniform full-tile fast path
            if (MODE == 0) {
                // rows row0..row0+7 contiguous; row0 % 8 == 0 -> 16B aligned
                float4 ns_lo = *reinterpret_cast<const float4*>(norm_s + row0);
                float4 ns_hi = *reinterpret_cast<const float4*>(norm_s + row0 + 4);
                float ns[8] = { ns_lo.x, ns_lo.y, ns_lo.z, ns_lo.w,
                                ns_hi.x, ns_hi.y, ns_hi.z, ns_hi.w };
#pragma unroll
                for (int r = 0; r < 8; ++r) {
                    float h = fmaxf(acc[r] + bcol, 0.0f);
                    hs_out[(size_t)(row0 + r) * 128 + col] = h * ns[r];
                }
            } else {
#pragma unroll
                for (int r = 0; r < 8; ++r) {
                    float h = fmaxf(acc[r] + bcol, 0.0f);
                    h_out[(size_t)(row0 + r) * 128 + col] = h;
                    colacc += h;
                }
            }
        } else {                                     // generic guarded tail
#pragma unroll
            for (int r = 0; r < 8; ++r) {
                int row = row0 + r;
                if (row < nN) {
                    float h = fmaxf(acc[r] + bcol, 0.0f);
                    if (MODE == 0) {
                        hs_out[(size_t)row * 128 + col] = h * norm_s[row];
                    } else {
                        h_out[(size_t)row * 128 + col] = h;
                        colacc += h;
                    }
                }
            }
        }
    }
    if (MODE == 1) atomicAdd(colsum + col, colacc);
}

// ---------------------------------------------------------------- MLP head
__global__ void gcn_head(const float* __restrict__ colsum, const float* __restrict__ Wl1,
                         const float* __restrict__ bl1, const float* __restrict__ Wl2,
                         const float* __restrict__ bl2, float* __restrict__ out, float invN) {
    __shared__ float emb[128];
    __shared__ float tj[64];
    int t = threadIdx.x;            // 128 threads
    float e = colsum[t] * invN;
    emb[t] = e;
    out[1 + t] = e;                 // graph_emb
    __syncthreads();
    if (t < 64) {
        float acc = bl1[t];
        for (int i = 0; i < 128; ++i) acc += emb[i] * Wl1[i * 64 + t];
        float he = acc > 0.0f ? acc : 0.0f;
        tj[t] = he * Wl2[t];
    }
    __syncthreads();
    if (t == 0) {
        float s = bl2[0];
        for (int j = 0; j < 64; ++j) s += tj[j];
        out[0] = 1.0f / (1.0f + expf(-s));   // pred
    }
}

// ---------------------------------------------------------------- launcher
extern "C" void kernel_launch(void* const* d_in, const int* in_sizes, int n_in,
                              void* d_out, int out_size, void* d_ws, size_t ws_size,
                              hipStream_t stream) {
    const float* coord_x = (const float*)d_in[0];
    const float* coord_y = (const float*)d_in[1];
    const float* W1  = (const float*)d_in[2];
    const float* b1  = (const float*)d_in[3];
    const float* Wn[4] = { (const float*)d_in[4], (const float*)d_in[6],
                           (const float*)d_in[8], (const float*)d_in[10] };
    const float* bn[4] = { (const float*)d_in[5], (const float*)d_in[7],
                           (const float*)d_in[9], (const float*)d_in[11] };
    const float* Wl1 = (const float*)d_in[12];
    const float* bl1 = (const float*)d_in[13];
    const float* Wl2 = (const float*)d_in[14];
    const float* bl2 = (const float*)d_in[15];
    const int*   src = (const int*)d_in[16];
    const int*   dst = (const int*)d_in[17];
    const int N = in_sizes[0];
    const int E = in_sizes[16];

    // workspace layout (floats): deg(2N) norms(2N) agg4(4N) hs0(4N) Wpad(512)
    // colsum(128) | hsA(128N) hsB(128N) agg(128N)   ~= 80 MB total
    float* ws      = (float*)d_ws;
    float* in_deg  = ws;
    float* out_deg = ws + (size_t)N;
    float* norm_s  = ws + 2 * (size_t)N;
    float* norm_d  = ws + 3 * (size_t)N;
    float* agg4    = ws + 4 * (size_t)N;
    float* hs0     = ws + 8 * (size_t)N;
    float* Wpad    = ws + 12 * (size_t)N;
    float* colsum  = Wpad + 512;
    float* hsA     = ws + 16 * (size_t)N;
    float* hsB     = hsA + 128 * (size_t)N;
    float* agg     = hsB + 128 * (size_t)N;

    float* out   = (float*)d_out;
    float* h_fin = out + 129;      // final h region [N,128]

    auto cdiv = [](int a, int b) { return (a + b - 1) / b; };
    const int nTiles = cdiv(N, 16);
    const int gemmBlocks = cdiv(nTiles, 8);

    // degrees + norms + layer-0 scaled features
    gcn_zero<<<cdiv(2 * N, 1024), 256, 0, stream>>>(in_deg, 2 * N);
    gcn_degree<<<cdiv(E, 256), 256, 0, stream>>>(src, dst, in_deg, out_deg, E);
    gcn_feat0<<<cdiv(N, 256), 256, 0, stream>>>(coord_x, coord_y, in_deg, out_deg,
                                                norm_s, norm_d, hs0, N);
    gcn_pad_w1<<<2, 256, 0, stream>>>(W1, Wpad);

    // layer 1 (K = 4, padded from 3)
    gcn_zero<<<cdiv(4 * N, 1024), 256, 0, stream>>>(agg4, 4 * N);
    gcn_scatter3<<<cdiv(E, 256), 256, 0, stream>>>(hs0, src, dst, agg4, E);
    gcn_gemm_wmma<4, 8, 0><<<gemmBlocks, 256, 0, stream>>>(
        agg4, Wpad, b1, norm_d, norm_s, hsA, nullptr, nullptr, N);

    // layers 2..5 (K = 128)
    float* hs_in = hsA;
    float* hs_nx = hsB;
    for (int l = 0; l < 4; ++l) {
        const bool fin = (l == 3);
        gcn_zero<<<cdiv(128 * N, 1024), 256, 0, stream>>>(agg, 128 * N);
        long long tot = (long long)E * 32;
        gcn_scatter128<<<(unsigned)((tot + 255) / 256), 256, 0, stream>>>(
            hs_in, src, dst, agg, E);
        if (fin) {
            gcn_zero<<<1, 64, 0, stream>>>(colsum, 128);
            gcn_gemm_wmma<128, 8, 1><<<gemmBlocks, 256, 0, stream>>>(
                agg, Wn[l], bn[l], norm_d, norm_s, nullptr, h_fin, colsum, N);
        } else {
            gcn_gemm_wmma<128, 8, 0><<<gemmBlocks, 256, 0, stream>>>(
                agg, Wn[l], bn[l], norm_d, norm_s, hs_nx, nullptr, nullptr, N);
        }
        float* tmp = hs_in; hs_in = hs_nx; hs_nx = tmp;
    }

    // mean-pool + MLP head
    gcn_head<<<1, 128, 0, stream>>>(colsum, Wl1, bl1, Wl2, bl2, out, 1.0f / (float)N);
}